// SAGEConvWithCV_49177375539675
// MI455X (gfx1250) — compile-verified
//
#include <hip/hip_runtime.h>
#include <hip/hip_bf16.h>

#define N_SRC   100000
#define N_DST   50000
#define N_EDGES 800000
#define D       64
#define D_OUT   64
#define ALPHA   0.1f

typedef __attribute__((ext_vector_type(2))) float v2f;
typedef __attribute__((ext_vector_type(8))) float v8f;

// ---------------------------------------------------------------------------
// Kernel 1: per-dst segment mean (atomic-free: dst_idx is sorted, so each wave
// binary-searches its edge range) fused with the control-variate blend.
// One wave (32 lanes) per dst node; lane l owns feature channels [2l, 2l+1],
// so each edge gather is one coalesced 256B row read of H_src.
// Writes h_neigh directly into the second output slot.
// ---------------------------------------------------------------------------
__global__ __launch_bounds__(128) void
seg_mean_blend_kernel(const float* __restrict__ H_src,
                      const float* __restrict__ HBar,
                      const int*   __restrict__ src_idx,
                      const int*   __restrict__ dst_idx,
                      float*       __restrict__ h_neigh)
{
    const int wave = (int)((blockIdx.x * blockDim.x + threadIdx.x) >> 5);
    const int lane = (int)(threadIdx.x & 31);
    if (wave >= N_DST) return;
    const int v = wave;

    // lower_bound(dst_idx, v)  -> start ; lower_bound(dst_idx, v+1) -> end
    int lo = 0, hi = N_EDGES;
    while (lo < hi) { int mid = (lo + hi) >> 1; if (dst_idx[mid] <  v) lo = mid + 1; else hi = mid; }
    const int start = lo;
    hi = N_EDGES;
    while (lo < hi) { int mid = (lo + hi) >> 1; if (dst_idx[mid] <= v) lo = mid + 1; else hi = mid; }
    const int end = lo;

    float ax = 0.0f, ay = 0.0f;
    for (int e = start; e < end; ++e) {
        const int s = src_idx[e];
        const v2f x = *(const v2f*)(H_src + (size_t)s * D + 2 * lane);
        ax += x.x;
        ay += x.y;
    }

    const float inv = 1.0f / fmaxf((float)(end - start), 1.0f);
    const v2f hb = *(const v2f*)(HBar + (size_t)v * D + 2 * lane);

    v2f hn;
    hn.x = (1.0f - ALPHA) * hb.x + ALPHA * (ax * inv);
    hn.y = (1.0f - ALPHA) * hb.y + ALPHA * (ay * inv);
    *(v2f*)(h_neigh + (size_t)v * D + 2 * lane) = hn;
}

// ---------------------------------------------------------------------------
// Kernel 2: h = relu(concat([H_dst, h_neigh]) @ W.T + b) via
// V_WMMA_F32_16X16X4_F32 (fp32 in / fp32 out, exact).
// One wave computes a 16-row x 64-col output tile: 4 accumulators (v8f each),
// K = 128 in 32 steps of 4.
//   A(m,k):  lane = m + 16*(k/2 within step), vgpr = k%2 -> contiguous float2
//   B(k,n) = W[n*128 + k]: same striping -> contiguous float2 per lane
//   C/D:     vgpr r, lanes 0-15 -> (M=r,   N=lane)
//                    lanes 16-31 -> (M=r+8, N=lane-16)
// ---------------------------------------------------------------------------
__global__ __launch_bounds__(128) void
sage_gemm_relu_wmma_kernel(const float* __restrict__ H_dst,
                           const float* __restrict__ h_neigh,
                           const float* __restrict__ W,
                           const float* __restrict__ b,
                           float*       __restrict__ out)
{
    const int wave = (int)((blockIdx.x * blockDim.x + threadIdx.x) >> 5);
    const int lane = (int)(threadIdx.x & 31);
    const int ntiles = N_DST / 16;              // 3125 exactly
    if (wave >= ntiles) return;

    const int r0    = wave * 16;
    const int m     = lane & 15;                // A-row / B-col / C-col index
    const int khalf = lane >> 4;                // K sub-pair selector

    const float* __restrict__ Xd = H_dst   + (size_t)(r0 + m) * D;
    const float* __restrict__ Xn = h_neigh + (size_t)(r0 + m) * D;

    // Seed accumulators with the bias (C(m,n) layout: N = lane&15 for all vgprs).
    v8f acc[4];
    #pragma unroll
    for (int t = 0; t < 4; ++t) {
        const float bias = b[t * 16 + m];
        #pragma unroll
        for (int r = 0; r < 8; ++r) acc[t][r] = bias;
    }

    // K = 0..63 : H_dst half of the concat
    #pragma unroll 4
    for (int ks = 0; ks < 16; ++ks) {
        const int k = ks * 4 + khalf * 2;
        const v2f a = *(const v2f*)(Xd + k);
        #pragma unroll
        for (int t = 0; t < 4; ++t) {
            const v2f bb = *(const v2f*)(W + (size_t)(t * 16 + m) * (2 * D) + k);
            acc[t] = __builtin_amdgcn_wmma_f32_16x16x4_f32(
                false, a, false, bb, (short)0, acc[t], false, false);
        }
    }
    // K = 64..127 : h_neigh half of the concat
    #pragma unroll 4
    for (int ks = 0; ks < 16; ++ks) {
        const int k = ks * 4 + khalf * 2;
        const v2f a = *(const v2f*)(Xn + k);
        #pragma unroll
        for (int t = 0; t < 4; ++t) {
            const v2f bb = *(const v2f*)(W + (size_t)(t * 16 + m) * (2 * D) + 64 + k);
            acc[t] = __builtin_amdgcn_wmma_f32_16x16x4_f32(
                false, a, false, bb, (short)0, acc[t], false, false);
        }
    }

    // ReLU + store. Row = r0 + r + 8*khalf, col = t*16 + (lane&15).
    #pragma unroll
    for (int t = 0; t < 4; ++t) {
        #pragma unroll
        for (int r = 0; r < 8; ++r) {
            const int row = r0 + r + 8 * khalf;
            const int col = t * 16 + m;
            out[(size_t)row * D_OUT + col] = fmaxf(acc[t][r], 0.0f);
        }
    }
}

extern "C" void kernel_launch(void* const* d_in, const int* in_sizes, int n_in,
                              void* d_out, int out_size, void* d_ws, size_t ws_size,
                              hipStream_t stream) {
    (void)in_sizes; (void)n_in; (void)out_size; (void)d_ws; (void)ws_size;

    const float* H_src   = (const float*)d_in[0];
    const float* H_dst   = (const float*)d_in[1];
    const float* HBar    = (const float*)d_in[2];
    const int*   src_idx = (const int*)d_in[3];
    const int*   dst_idx = (const int*)d_in[4];
    const float* W       = (const float*)d_in[5];
    const float* b       = (const float*)d_in[6];

    float* h       = (float*)d_out;                       // output 0: [N_DST, 64]
    float* h_neigh = h + (size_t)N_DST * D_OUT;           // output 1: [N_DST, 64]

    // 4 waves per 128-thread block; one wave per dst node.
    const int waves1  = N_DST;
    const int blocks1 = (waves1 + 3) / 4;
    seg_mean_blend_kernel<<<blocks1, 128, 0, stream>>>(H_src, HBar, src_idx, dst_idx, h_neigh);

    // One wave per 16-row output tile.
    const int tiles   = N_DST / 16;                       // 3125
    const int blocks2 = (tiles + 3) / 4;
    sage_gemm_relu_wmma_kernel<<<blocks2, 128, 0, stream>>>(H_dst, h_neigh, W, b, h);
}